// MambaLayer_75479755260145
// MI455X (gfx1250) — compile-verified
//
#include <hip/hip_runtime.h>
#include <hip/hip_bf16.h>
#include <math.h>

// ---- problem constants (match reference) ----
#define D_MODEL 1024
#define D_STATE 16
#define D_INNER 2048
#define DT_RANK 64
#define D_CONV  4
#define B_SZ    2
#define L_SEQ   2048
#define M_TOK   (B_SZ * L_SEQ)           // 4096 tokens
#define XPROJ_N (DT_RANK + 2 * D_STATE)  // 96
#define GEMM_KC 32                        // K-chunk staged per TDM transfer

typedef __attribute__((ext_vector_type(2))) float        v2f;
typedef __attribute__((ext_vector_type(8))) float        v8f;
typedef __attribute__((ext_vector_type(4))) unsigned int v4u;
typedef __attribute__((ext_vector_type(8))) int          v8i_t;
typedef __attribute__((ext_vector_type(4))) int          v4i_t;

__device__ __forceinline__ float silu_f(float v) {
    return v / (1.f + __expf(-v));
}

// ---------------------------------------------------------------------------
// TDM: stage a (rows x GEMM_KC) fp32 tile of row-major W (leading dim ldw)
// into LDS at byte offset lds_byte, with 2-dword padding after every 32
// dwords (-> LDS row stride = GEMM_KC+2 floats, kills ds bank conflicts).
// D# layout per CDNA5 ISA (group0 128b, group1 256b).
// clang-23 toolchain: 6-arg builtin (g0, g1, g2, g3, extra, cpol).
// ---------------------------------------------------------------------------
__device__ __forceinline__ void tdm_load_tile(const float* gsrc, unsigned int lds_byte,
                                              int rows, int ldw_elems) {
    unsigned long long ga = (unsigned long long)(uintptr_t)gsrc;
    v4u g0;
    g0[0] = 1u;                                   // count=1, user descriptor
    g0[1] = lds_byte;                             // lds_addr
    g0[2] = (unsigned int)ga;                     // global_addr[31:0]
    g0[3] = (unsigned int)(ga >> 32) | (2u << 30);// global_addr[56:32] | type=2
    v8i_t g1;
    g1[0] = (int)((2u << 16)        // data_size = 2 -> 4 bytes
                | (1u << 20)        // pad_enable
                | (4u << 22)        // pad_interval: 32 dwords
                | (1u << 25));      // pad_amount:   2 dwords
    g1[1] = (int)((unsigned)GEMM_KC << 16);       // tensor_dim0 (low 16)
    g1[2] = (int)((unsigned)rows << 16);          // tensor_dim1 (low 16)
    g1[3] = (int)((unsigned)GEMM_KC << 16);       // tile_dim0
    g1[4] = rows;                                 // tile_dim1 (tile_dim2 = 0)
    g1[5] = ldw_elems;                            // tensor_dim0_stride[31:0]
    g1[6] = 0;
    g1[7] = 0;
    v4i_t z4;
    z4[0] = 0; z4[1] = 0; z4[2] = 0; z4[3] = 0;
    v8i_t z8;
    z8[0] = 0; z8[1] = 0; z8[2] = 0; z8[3] = 0;
    z8[4] = 0; z8[5] = 0; z8[6] = 0; z8[7] = 0;
    __builtin_amdgcn_tensor_load_to_lds(g0, g1, z4, z4, z8, 0);
}

// ---------------------------------------------------------------------------
// Kernel 1: LayerNorm, one block per token row (1024 elements)
// ---------------------------------------------------------------------------
__global__ __launch_bounds__(256)
void ln_kernel(const float* __restrict__ x, const float* __restrict__ g,
               const float* __restrict__ b, float* __restrict__ h) {
    __shared__ float ssum[256];
    __shared__ float ssq[256];
    const int m = blockIdx.x;
    const int t = threadIdx.x;
    float s = 0.f, q = 0.f;
    for (int k = t; k < D_MODEL; k += 256) {
        float v = x[(size_t)m * D_MODEL + k];
        s += v; q += v * v;
    }
    ssum[t] = s; ssq[t] = q;
    __syncthreads();
    for (int off = 128; off > 0; off >>= 1) {
        if (t < off) { ssum[t] += ssum[t + off]; ssq[t] += ssq[t + off]; }
        __syncthreads();
    }
    const float mu  = ssum[0] * (1.f / D_MODEL);
    const float var = ssq[0] * (1.f / D_MODEL) - mu * mu;
    const float inv = rsqrtf(var + 1e-5f);
    for (int k = t; k < D_MODEL; k += 256) {
        float v = x[(size_t)m * D_MODEL + k];
        h[(size_t)m * D_MODEL + k] = (v - mu) * inv * g[k] + b[k];
    }
}

// ---------------------------------------------------------------------------
// fp32 WMMA GEMM  C[m,n] = sum_k A[m,k] * W[n,k]  (+ fused epilogue)
//   block = 128 threads = 4 waves sharing one N strip (NT*16 wide);
//   W tiles staged global->LDS by the Tensor Data Mover, double buffered,
//   synchronized with s_wait_tensorcnt + workgroup barrier;
//   A fragments of each K-chunk loaded into registers just before its wmma loop.
//   MODE 0: plain store
//   MODE 1: softplus(val + extra[n])          (dt_proj bias + softplus)
//   MODE 2: val + extra[m*ldc + n]            (residual add)
// ---------------------------------------------------------------------------
template <int NT, int MODE>
__global__ __launch_bounds__(128)
void gemm_wmma_kernel(const float* __restrict__ A, int lda,
                      const float* __restrict__ W, int ldw,
                      float* __restrict__ C, int ldc, int K,
                      const float* __restrict__ extra) {
    constexpr int ROWS = NT * 16;
    __shared__ float sW[2][ROWS][GEMM_KC + 2];   // +2: matches TDM pad

    const int wave = threadIdx.x >> 5;
    const int lane = threadIdx.x & 31;
    const int half = lane >> 4;   // 0: K pair {0,1}/rows 0-7 ; 1: K pair {2,3}/rows 8-15
    const int r    = lane & 15;

    const int m0 = (blockIdx.y * 4 + wave) * 16;
    const int n0 = blockIdx.x * ROWS;

    v8f acc[NT];
#pragma unroll
    for (int j = 0; j < NT; ++j)
#pragma unroll
        for (int e = 0; e < 8; ++e) acc[j][e] = 0.f;

    const float* ap0   = A + (size_t)(m0 + r) * lda + 2 * half;
    const float* wbase = W + (size_t)n0 * ldw;
    const int nchunk = K / GEMM_KC;

    if (wave == 0)
        tdm_load_tile(wbase, (unsigned int)(uintptr_t)&sW[0][0][0], ROWS, ldw);

    for (int c = 0; c < nchunk; ++c) {
        const int buf = c & 1;
        if (wave == 0) __builtin_amdgcn_s_wait_tensorcnt(0);  // chunk c landed
        __syncthreads();
        if (wave == 0 && c + 1 < nchunk)                      // prefetch chunk c+1
            tdm_load_tile(wbase + (size_t)(c + 1) * GEMM_KC,
                          (unsigned int)(uintptr_t)&sW[buf ^ 1][0][0], ROWS, ldw);

        // register-prefetch all A fragments of this chunk (8 x b64 per lane)
        v2f areg[GEMM_KC / 4];
        const float* ac = ap0 + (size_t)c * GEMM_KC;
#pragma unroll
        for (int s = 0; s < GEMM_KC / 4; ++s)
            areg[s] = *(const v2f*)(ac + 4 * s);

#pragma unroll
        for (int s = 0; s < GEMM_KC / 4; ++s) {
            const int kk = 4 * s;
#pragma unroll
            for (int j = 0; j < NT; ++j) {
                // B fragment 4x16 of W^T, served from LDS (ds_load_b64)
                v2f bm = *(const v2f*)&sW[buf][j * 16 + r][kk + 2 * half];
                acc[j] = __builtin_amdgcn_wmma_f32_16x16x4_f32(
                    false, areg[s], false, bm, (short)0, acc[j], false, false);
            }
        }
        // no trailing barrier: top-of-loop barrier orders buffer reuse
    }

#pragma unroll
    for (int j = 0; j < NT; ++j) {
        const int n = n0 + j * 16 + r;
#pragma unroll
        for (int v = 0; v < 8; ++v) {
            const int m = m0 + v + 8 * half;   // C/D layout: VGPR v -> row v (+8 hi half)
            float val = acc[j][v];
            if (MODE == 1) {
                float t = val + extra[n];
                val = (t > 20.f) ? t : log1pf(__expf(t));   // softplus
            } else if (MODE == 2) {
                val += extra[(size_t)m * ldc + n];          // residual
            }
            C[(size_t)m * ldc + n] = val;
        }
    }
}

// ---------------------------------------------------------------------------
// Kernel 3: causal depthwise conv (K=4) + SiLU.  xin = xz[..., :D_INNER]
// ---------------------------------------------------------------------------
__global__ __launch_bounds__(256)
void conv_silu_kernel(const float* __restrict__ xz, const float* __restrict__ w,
                      const float* __restrict__ bias, float* __restrict__ xc) {
    const int d   = blockIdx.x * 256 + threadIdx.x;   // 0..D_INNER-1
    const int row = blockIdx.y;                       // 0..M_TOK-1
    const int b   = row / L_SEQ;
    const int l   = row % L_SEQ;
    float acc = bias[d];
#pragma unroll
    for (int k = 0; k < D_CONV; ++k) {
        const int lk = l - (D_CONV - 1) + k;
        if (lk >= 0)
            acc += xz[(size_t)(b * L_SEQ + lk) * (2 * D_INNER) + d] * w[d * D_CONV + k];
    }
    xc[(size_t)row * D_INNER + d] = silu_f(acc);
}

// ---------------------------------------------------------------------------
// Kernel 6: selective scan. One thread per (batch, channel d); 16-state
// recurrence in registers; B/C rows (shared by all d) staged via LDS in
// 64-step chunks. Fuses y = scan + u*D, then * silu(z).
// ---------------------------------------------------------------------------
#define SCAN_CHUNK 64
__global__ __launch_bounds__(256)
void scan_kernel(const float* __restrict__ xc, const float* __restrict__ dt,
                 const float* __restrict__ dbl, const float* __restrict__ A_log,
                 const float* __restrict__ Dv, const float* __restrict__ xz,
                 float* __restrict__ ya) {
    __shared__ float sBC[SCAN_CHUNK][2 * D_STATE];  // [i][0:16]=B, [i][16:32]=C

    const int d = blockIdx.x * 256 + threadIdx.x;   // channel
    const int b = blockIdx.y;                       // batch

    float Arow[D_STATE];
#pragma unroll
    for (int n = 0; n < D_STATE; ++n)
        Arow[n] = -__expf(A_log[(size_t)d * D_STATE + n]);
    const float Dd = Dv[d];

    float h[D_STATE];
#pragma unroll
    for (int n = 0; n < D_STATE; ++n) h[n] = 0.f;

    for (int l0 = 0; l0 < L_SEQ; l0 += SCAN_CHUNK) {
        __syncthreads();
        for (int j = threadIdx.x; j < SCAN_CHUNK * 2 * D_STATE; j += 256) {
            const int i = j >> 5;
            const int c = j & 31;
            sBC[i][c] = dbl[(size_t)(b * L_SEQ + l0 + i) * XPROJ_N + DT_RANK + c];
        }
        __syncthreads();

        for (int i = 0; i < SCAN_CHUNK; ++i) {
            const size_t row = (size_t)(b * L_SEQ + l0 + i);
            const float dtv = dt[row * D_INNER + d];
            const float u   = xc[row * D_INNER + d];
            const float du  = dtv * u;
            float y = 0.f;
#pragma unroll
            for (int n = 0; n < D_STATE; ++n) {
                const float dA = __expf(dtv * Arow[n]);
                h[n] = dA * h[n] + du * sBC[i][n];
                y += h[n] * sBC[i][D_STATE + n];
            }
            y += u * Dd;
            const float z = xz[row * (2 * D_INNER) + D_INNER + d];
            ya[row * D_INNER + d] = y * silu_f(z);
        }
    }
}

// ---------------------------------------------------------------------------
// launcher
// ---------------------------------------------------------------------------
extern "C" void kernel_launch(void* const* d_in, const int* in_sizes, int n_in,
                              void* d_out, int out_size, void* d_ws, size_t ws_size,
                              hipStream_t stream) {
    (void)in_sizes; (void)n_in; (void)out_size; (void)ws_size;

    const float* x         = (const float*)d_in[0];   // (B,L,1024)
    const float* ln_gamma  = (const float*)d_in[1];
    const float* ln_beta   = (const float*)d_in[2];
    const float* in_proj_w = (const float*)d_in[3];   // (4096,1024)
    const float* conv_w    = (const float*)d_in[4];   // (2048,4)
    const float* conv_b    = (const float*)d_in[5];
    const float* x_proj_w  = (const float*)d_in[6];   // (96,2048)
    const float* dt_proj_w = (const float*)d_in[7];   // (2048,64)
    const float* dt_proj_b = (const float*)d_in[8];
    const float* A_log     = (const float*)d_in[9];   // (2048,16)
    const float* Dvec      = (const float*)d_in[10];  // (2048,)
    const float* out_proj_w= (const float*)d_in[11];  // (1024,2048)
    float*       out       = (float*)d_out;

    // workspace layout (floats)
    float* ws   = (float*)d_ws;
    float* h    = ws;                                       // 4096*1024
    float* xz   = h    + (size_t)M_TOK * D_MODEL;           // 4096*4096
    float* xc   = xz   + (size_t)M_TOK * 2 * D_INNER;       // 4096*2048
    float* dbl  = xc   + (size_t)M_TOK * D_INNER;           // 4096*96
    float* dtb  = dbl  + (size_t)M_TOK * XPROJ_N;           // 4096*2048
    float* ya   = dtb  + (size_t)M_TOK * D_INNER;           // 4096*2048

    // 1) LayerNorm
    ln_kernel<<<M_TOK, 256, 0, stream>>>(x, ln_gamma, ln_beta, h);

    // 2) in_proj: xz = h @ in_proj_w^T   (M=4096, N=4096, K=1024)
    gemm_wmma_kernel<4, 0><<<dim3((2 * D_INNER) / 64, M_TOK / 64), 128, 0, stream>>>(
        h, D_MODEL, in_proj_w, D_MODEL, xz, 2 * D_INNER, D_MODEL, nullptr);

    // 3) causal depthwise conv + silu on xin = xz[..., :2048]
    conv_silu_kernel<<<dim3(D_INNER / 256, M_TOK), 256, 0, stream>>>(
        xz, conv_w, conv_b, xc);

    // 4) x_proj: dbl = xc @ x_proj_w^T   (M=4096, N=96, K=2048)
    gemm_wmma_kernel<2, 0><<<dim3(XPROJ_N / 32, M_TOK / 64), 128, 0, stream>>>(
        xc, D_INNER, x_proj_w, D_INNER, dbl, XPROJ_N, D_INNER, nullptr);

    // 5) dt_proj + bias + softplus: dtb = softplus(dbl[:, :64] @ dt_proj_w^T + b)
    gemm_wmma_kernel<4, 1><<<dim3(D_INNER / 64, M_TOK / 64), 128, 0, stream>>>(
        dbl, XPROJ_N, dt_proj_w, DT_RANK, dtb, D_INNER, DT_RANK, dt_proj_b);

    // 6) selective scan + D skip + silu(z) gate
    scan_kernel<<<dim3(D_INNER / 256, B_SZ), 256, 0, stream>>>(
        xc, dtb, dbl, A_log, Dvec, xz, ya);

    // 7) out_proj + residual: out = x + ya @ out_proj_w^T  (M=4096, N=1024, K=2048)
    gemm_wmma_kernel<4, 2><<<dim3(D_MODEL / 64, M_TOK / 64), 128, 0, stream>>>(
        ya, D_INNER, out_proj_w, D_INNER, out, D_MODEL, D_INNER, x);
}